// agg_NodeGCN1_16226386444400
// MI455X (gfx1250) — compile-verified
//
#include <hip/hip_runtime.h>
#include <hip/hip_bf16.h>

typedef __attribute__((ext_vector_type(2))) float v2f;
typedef __attribute__((ext_vector_type(8))) float v8f;

#define FDIM 128

// ---------------- degree / norm ----------------

__global__ void init_deg_kernel(float* __restrict__ deg, int n) {
    int i = blockIdx.x * blockDim.x + threadIdx.x;
    if (i < n) deg[i] = 1.0f;               // self-loop weight
}

__global__ void scatter_deg_kernel(const long long* __restrict__ col,
                                   const float* __restrict__ ew,
                                   float* __restrict__ deg, int e) {
    int i = blockIdx.x * blockDim.x + threadIdx.x;
    if (i < e)
        __hip_atomic_fetch_add(&deg[(int)col[i]], ew[i],
                               __ATOMIC_RELAXED, __HIP_MEMORY_SCOPE_AGENT);
}

__global__ void deg_to_dis_kernel(float* __restrict__ d, int n) {
    int i = blockIdx.x * blockDim.x + threadIdx.x;
    if (i < n) {
        float v = d[i];
        d[i] = (v > 0.0f) ? rsqrtf(v) : 0.0f;
    }
}

__global__ void norm_kernel(const long long* __restrict__ row,
                            const long long* __restrict__ col,
                            const float* __restrict__ ew,
                            const float* __restrict__ dis,
                            float* __restrict__ norm, int e) {
    int i = blockIdx.x * blockDim.x + threadIdx.x;
    if (i < e)
        norm[i] = dis[(int)row[i]] * ew[i] * dis[(int)col[i]];
}

// ---------------- fp32 WMMA GEMM: H = A(N x 128) * W(128 x 128) ----------------
// One wave -> one 16x16 output tile via V_WMMA_F32_16X16X4_F32 (exact fp32).
// Block = 256 threads = 8 waves -> 16 rows x 128 cols.

#define LDSA_STRIDE 132   // pad to avoid LDS bank conflicts (bank = 4m+k)

__global__ __launch_bounds__(256) void gemm128_wmma_kernel(const float* __restrict__ A,
                                                           const float* __restrict__ W,
                                                           float* __restrict__ H, int n) {
    __shared__ float As[16 * LDSA_STRIDE];
    const int tid  = threadIdx.x;
    const int wave = tid >> 5;
    const int lane = tid & 31;
    const int row0 = blockIdx.x * 16;

    // Stage 16x128 A-tile into LDS (512 float4 transfers, 2 per thread)
    for (int idx = tid; idx < 512; idx += 256) {
        int r  = idx >> 5;        // row within tile
        int cv = idx & 31;        // float4 column
        float4 v = make_float4(0.f, 0.f, 0.f, 0.f);
        if (row0 + r < n)
            v = *(const float4*)(A + (size_t)(row0 + r) * FDIM + cv * 4);
        *(float4*)(&As[r * LDSA_STRIDE + cv * 4]) = v;
    }
    __syncthreads();

    const int col0 = wave * 16;
    const int m    = lane & 15;
    const int kb   = (lane >> 4) * 2;
    const int nn   = lane & 15;

    v8f acc = {};
    for (int k0 = 0; k0 < FDIM; k0 += 4) {
        v2f a, b;
        a.x = As[m * LDSA_STRIDE + k0 + kb];
        a.y = As[m * LDSA_STRIDE + k0 + kb + 1];
        b.x = W[(k0 + kb) * FDIM + col0 + nn];
        b.y = W[(k0 + kb + 1) * FDIM + col0 + nn];
        acc = __builtin_amdgcn_wmma_f32_16x16x4_f32(false, a, false, b,
                                                    (short)0, acc, false, false);
    }

    const int mbase = (lane >> 4) * 8;
#pragma unroll
    for (int i = 0; i < 8; ++i) {
        int r = row0 + mbase + i;
        if (r < n) H[(size_t)r * FDIM + col0 + nn] = acc[i];
    }
}

// ---------------- aggregation ----------------

// o[i][f] = b[f] + h[i][f] * dis[i]^2   (bias + self-loop contribution)
__global__ void agg_init_kernel(const float* __restrict__ h,
                                const float* __restrict__ dis,
                                const float* __restrict__ b,
                                float* __restrict__ o, int n) {
    int i = blockIdx.x * blockDim.x + threadIdx.x;
    if (i < n * FDIM) {
        int node = i >> 7, f = i & 127;
        float d = dis[node];
        o[i] = b[f] + h[i] * d * d;
    }
}

// o[col] += h[row] * norm  (128 threads per edge, 2 edges per block)
__global__ __launch_bounds__(256) void agg_edges_kernel(const long long* __restrict__ row,
                                                        const long long* __restrict__ col,
                                                        const float* __restrict__ norm,
                                                        const float* __restrict__ h,
                                                        float* __restrict__ o, int e) {
    int ei = blockIdx.x * 2 + (threadIdx.x >> 7);
    if (ei >= e) return;
    int f = threadIdx.x & 127;
    int r = (int)row[ei], c = (int)col[ei];
    float w = norm[ei];
    __hip_atomic_fetch_add(&o[c * FDIM + f], h[r * FDIM + f] * w,
                           __ATOMIC_RELAXED, __HIP_MEMORY_SCOPE_AGENT);
}

// ---------------- batch norm + relu ----------------

__global__ void zero_stats_kernel(float* __restrict__ stats) {
    stats[threadIdx.x] = 0.0f;   // 256 entries: sum[128], sumsq[128]
}

__global__ __launch_bounds__(256) void bn_stats_kernel(const float* __restrict__ o, int n,
                                                       float* __restrict__ stats) {
    const int c    = threadIdx.x & 127;
    const int half = threadIdx.x >> 7;
    int base = blockIdx.x * 256;
    int end  = base + 256; if (end > n) end = n;
    float s = 0.f, s2 = 0.f;
    for (int r = base + half; r < end; r += 2) {
        float v = o[(size_t)r * FDIM + c];
        s += v; s2 += v * v;
    }
    __shared__ float sh[256], sh2[256];
    sh[threadIdx.x] = s; sh2[threadIdx.x] = s2;
    __syncthreads();
    if (half == 0) {
        float ts  = sh[c] + sh[c + 128];
        float ts2 = sh2[c] + sh2[c + 128];
        __hip_atomic_fetch_add(&stats[c], ts, __ATOMIC_RELAXED, __HIP_MEMORY_SCOPE_AGENT);
        __hip_atomic_fetch_add(&stats[c + 128], ts2, __ATOMIC_RELAXED, __HIP_MEMORY_SCOPE_AGENT);
    }
}

// stats[0:128] -> scale, stats[128:256] -> shift  (in place)
__global__ void bn_finalize_kernel(float* __restrict__ stats,
                                   const float* __restrict__ g,
                                   const float* __restrict__ be, float inv_n) {
    int c = threadIdx.x;
    float mean = stats[c] * inv_n;
    float var  = stats[c + 128] * inv_n - mean * mean;
    float sc   = g[c] * rsqrtf(var + 1e-5f);
    stats[c]       = sc;
    stats[c + 128] = be[c] - mean * sc;
}

__global__ void bn_apply_relu_kernel(float* __restrict__ o,
                                     const float* __restrict__ stats, int n) {
    int i = blockIdx.x * blockDim.x + threadIdx.x;
    if (i < n * FDIM) {
        int c = i & 127;
        float v = o[i] * stats[c] + stats[c + 128];
        o[i] = v > 0.f ? v : 0.f;
    }
}

// ---------------- final linear: out = [o1|o2|o3] @ Wlin + blin ----------------

__global__ void final_linear_kernel(const float* __restrict__ o1,
                                    const float* __restrict__ o2,
                                    const float* __restrict__ o3,
                                    const float* __restrict__ Wlin,
                                    const float* __restrict__ blin,
                                    float* __restrict__ out, int n) {
    int idx  = blockIdx.x * blockDim.x + threadIdx.x;
    int node = idx / 10;
    int c    = idx - node * 10;
    if (node >= n) return;
    float acc = blin[c];
    const float* r1 = o1 + (size_t)node * FDIM;
    const float* r2 = o2 + (size_t)node * FDIM;
    const float* r3 = o3 + (size_t)node * FDIM;
    for (int k = 0; k < FDIM; ++k) acc += r1[k] * Wlin[k * 10 + c];
    for (int k = 0; k < FDIM; ++k) acc += r2[k] * Wlin[(FDIM + k) * 10 + c];
    for (int k = 0; k < FDIM; ++k) acc += r3[k] * Wlin[(2 * FDIM + k) * 10 + c];
    out[(size_t)node * 10 + c] = acc;
}

// ---------------- host side ----------------

static void run_layer(const float* in, const float* W, const float* b,
                      const float* g, const float* be,
                      float* h, float* o, const float* dis,
                      const long long* rowp, const long long* colp,
                      const float* norm, float* stats,
                      int N, int E, hipStream_t stream) {
    int elem = N * FDIM;
    gemm128_wmma_kernel<<<(N + 15) / 16, 256, 0, stream>>>(in, W, h, N);
    agg_init_kernel<<<(elem + 255) / 256, 256, 0, stream>>>(h, dis, b, o, N);
    agg_edges_kernel<<<(E + 1) / 2, 256, 0, stream>>>(rowp, colp, norm, h, o, E);
    zero_stats_kernel<<<1, 256, 0, stream>>>(stats);
    bn_stats_kernel<<<(N + 255) / 256, 256, 0, stream>>>(o, N, stats);
    bn_finalize_kernel<<<1, 128, 0, stream>>>(stats, g, be, 1.0f / (float)N);
    bn_apply_relu_kernel<<<(elem + 255) / 256, 256, 0, stream>>>(o, stats, N);
}

extern "C" void kernel_launch(void* const* d_in, const int* in_sizes, int n_in,
                              void* d_out, int out_size, void* d_ws, size_t ws_size,
                              hipStream_t stream) {
    const float*     x    = (const float*)d_in[0];
    const long long* ei   = (const long long*)d_in[1];
    const float*     ew   = (const float*)d_in[2];
    const float*     W1   = (const float*)d_in[3];
    const float*     b1   = (const float*)d_in[4];
    const float*     g1   = (const float*)d_in[5];
    const float*     be1  = (const float*)d_in[6];
    const float*     W2   = (const float*)d_in[7];
    const float*     b2   = (const float*)d_in[8];
    const float*     g2   = (const float*)d_in[9];
    const float*     be2  = (const float*)d_in[10];
    const float*     W3   = (const float*)d_in[11];
    const float*     b3   = (const float*)d_in[12];
    const float*     g3   = (const float*)d_in[13];
    const float*     be3  = (const float*)d_in[14];
    const float*     Wlin = (const float*)d_in[15];
    const float*     blin = (const float*)d_in[16];
    float*           out  = (float*)d_out;

    const int N = in_sizes[0] / FDIM;     // 100000
    const int E = in_sizes[2];            // 1600000

    // workspace layout (floats)
    float* ws    = (float*)d_ws;
    size_t off   = 0;
    float* dis   = ws + off; off += (size_t)N;
    float* norm  = ws + off; off += (size_t)E;
    float* h     = ws + off; off += (size_t)N * FDIM;
    float* o1    = ws + off; off += (size_t)N * FDIM;
    float* o2    = ws + off; off += (size_t)N * FDIM;
    float* o3    = ws + off; off += (size_t)N * FDIM;
    float* stats = ws + off;

    const long long* rowp = ei;       // source j
    const long long* colp = ei + E;   // target i

    // gcn_norm
    init_deg_kernel<<<(N + 255) / 256, 256, 0, stream>>>(dis, N);
    scatter_deg_kernel<<<(E + 255) / 256, 256, 0, stream>>>(colp, ew, dis, E);
    deg_to_dis_kernel<<<(N + 255) / 256, 256, 0, stream>>>(dis, N);
    norm_kernel<<<(E + 255) / 256, 256, 0, stream>>>(rowp, colp, ew, dis, norm, E);

    // three GCN layers
    run_layer(x,  W1, b1, g1, be1, h, o1, dis, rowp, colp, norm, stats, N, E, stream);
    run_layer(o1, W2, b2, g2, be2, h, o2, dis, rowp, colp, norm, stats, N, E, stream);
    run_layer(o2, W3, b3, g3, be3, h, o3, dis, rowp, colp, norm, stats, N, E, stream);

    // classifier head
    final_linear_kernel<<<(N * 10 + 255) / 256, 256, 0, stream>>>(o1, o2, o3, Wlin, blin, out, N);
}